// End2EndRVFixedOutput_TorchScript_90933047590941
// MI455X (gfx1250) — compile-verified
//
#include <hip/hip_runtime.h>

// ---------------------------------------------------------------------------
// Batched YOLO NMS for MI455X (gfx1250, wave32).
//   in : x  (16, 25200, 85) f32
//   out: det (16, 300, 6)   f32   [x1,y1,x2,y2,score,class] (zeroed if invalid)
// ---------------------------------------------------------------------------

#define B_IMG   16
#define N_ANC   25200
#define ROW     85
#define NC      80
#define KTOP    1000
#define KPAD    1024
#define MAXDET  300
#define CONF    0.25f
#define IOU_T   0.45f
#define NEGV    -1000000000.0f
#define MAXWH   4096.0f
#define APB     128   // anchors per block in stage 1

// ---- CDNA5 async global->LDS path (Ch.8 async tensor ops), with fallback ---
#if defined(__has_builtin)
#if __has_builtin(__builtin_amdgcn_global_load_async_to_lds_b128) && \
    __has_builtin(__builtin_amdgcn_s_wait_asynccnt)
#define HAS_ASYNC 1
#endif
#endif
#ifndef HAS_ASYNC
#define HAS_ASYNC 0
#endif

typedef int v4i __attribute__((vector_size(16)));   // matches builtin pointee

__device__ __forceinline__ void g2l_b128(void* lds_dst, const void* g_src) {
#if HAS_ASYNC
  typedef __attribute__((address_space(1))) v4i GV;  // global (prints __device__)
  typedef __attribute__((address_space(3))) v4i LV;  // LDS
  // 16B per lane, wave32 -> 512B per instruction, tracked on ASYNCcnt.
  __builtin_amdgcn_global_load_async_to_lds_b128((GV*)g_src, (LV*)lds_dst, 0, 0);
#else
  *(float4*)lds_dst = *(const float4*)g_src;
#endif
}

__device__ __forceinline__ void g2l_wait() {
#if HAS_ASYNC
  __builtin_amdgcn_s_wait_asynccnt(0);
#endif
}

// ---------------------------------------------------------------------------
// Stage 1: per-anchor score/class/key.  HBM-bound: 137 MB -> ~6us @ 23.3 TB/s.
// Tile of 128 rows x 85 f32 is contiguous -> async DMA to LDS, then each
// thread reduces one row out of LDS (stride 85 is odd -> conflict-free banks).
// Key = sortable(score)<<32 | (0xFFFFFFFF - idx): matches jax top_k ordering
// (descending score, ascending index on ties) as a plain u64 descending sort.
// ---------------------------------------------------------------------------
__global__ __launch_bounds__(APB) void score_kernel(
    const float* __restrict__ x,
    unsigned long long* __restrict__ keys,
    int* __restrict__ clsArr) {
  __shared__ float tile[APB * ROW];  // 43,520 B
  const int tid  = threadIdx.x;
  const int nblk = (N_ANC + APB - 1) / APB;            // 197
  const int b    = blockIdx.x / nblk;
  const int a0   = (blockIdx.x % nblk) * APB;
  const int rows = min(APB, N_ANC - a0);               // 128 or 112 (mult of 4)
  const float* src = x + ((size_t)b * N_ANC + a0) * ROW;

  const int chunks = rows * ROW / 4;                   // 16B chunks, exact
  for (int c = tid; c < chunks; c += APB)
    g2l_b128((char*)tile + (size_t)c * 16, (const char*)src + (size_t)c * 16);
  g2l_wait();
  __syncthreads();

  if (tid < rows) {
    const float* r = tile + tid * ROW;
    float obj = r[4];
    float mx  = r[5];
    int   mc  = 0;
#pragma unroll 4
    for (int c = 1; c < NC; ++c) {
      float v = r[5 + c];
      if (v > mx) { mx = v; mc = c; }                  // strict > : first max
    }
    float score = obj * mx;
    float sf    = (score > CONF) ? score : NEGV;       // threshold like ref
    unsigned ub = __float_as_uint(sf);
    ub = (ub & 0x80000000u) ? ~ub : (ub | 0x80000000u);// order-preserving map
    int n = a0 + tid;
    size_t o = (size_t)b * N_ANC + n;
    keys[o]   = ((unsigned long long)ub << 32) |
                (unsigned long long)(0xFFFFFFFFu - (unsigned)n);
    clsArr[o] = mc;
  }
}

// ---------------------------------------------------------------------------
// Stage 2: exact top-1000 per image via 64-bit radix select (8x 256-bin LDS
// histogram passes; keys are distinct so the 1000th key is unique), then
// compact + 1024-wide bitonic sort (descending).  One WG per image.
// ---------------------------------------------------------------------------
__global__ __launch_bounds__(1024) void topk_kernel(
    const unsigned long long* __restrict__ keys,
    unsigned long long* __restrict__ topkeys) {
  __shared__ unsigned s_hist[256];
  __shared__ unsigned long long s_prefix;
  __shared__ int s_rem;
  __shared__ int s_cnt;
  __shared__ unsigned long long sbuf[KPAD];
  const int tid = threadIdx.x;
  const int b   = blockIdx.x;
  const unsigned long long* kin = keys + (size_t)b * N_ANC;

  if (tid == 0) { s_prefix = 0ull; s_rem = KTOP; s_cnt = 0; }

  for (int d = 7; d >= 0; --d) {
    if (tid < 256) s_hist[tid] = 0u;
    __syncthreads();
    const unsigned long long pfx = s_prefix;
    const int shift = d * 8;
    for (int i = tid; i < N_ANC; i += 1024) {
      unsigned long long k = kin[i];
      bool match = (d == 7) || ((k >> (shift + 8)) == pfx);
      if (match) atomicAdd(&s_hist[(unsigned)((k >> shift) & 0xFFull)], 1u);
    }
    __syncthreads();
    if (tid == 0) {
      int rem = s_rem, acc = 0, v = 255;
      for (; v > 0; --v) {
        int h = (int)s_hist[v];
        if (acc + h >= rem) break;
        acc += h;
      }
      s_rem    = rem - acc;
      s_prefix = (pfx << 8) | (unsigned long long)(unsigned)v;
    }
    __syncthreads();
  }
  const unsigned long long thr = s_prefix;  // exact 1000th-largest key

  if (tid >= KTOP) sbuf[tid] = 0ull;        // pad slots [1000,1024)
  for (int i = tid; i < N_ANC; i += 1024) {
    unsigned long long k = kin[i];
    if (k >= thr) {
      int p = atomicAdd(&s_cnt, 1);
      if (p < KTOP) sbuf[p] = k;            // exactly KTOP pass (distinct keys)
    }
  }
  __syncthreads();

  // bitonic sort, descending, 1024 elems / 1024 threads (55 stages)
  for (int k2 = 2; k2 <= KPAD; k2 <<= 1) {
    for (int j = k2 >> 1; j > 0; j >>= 1) {
      int p = tid ^ j;
      if (p > tid) {
        unsigned long long a = sbuf[tid], c = sbuf[p];
        bool descBlk = ((tid & k2) == 0);
        bool sw = descBlk ? (a < c) : (a > c);
        if (sw) { sbuf[tid] = c; sbuf[p] = a; }
      }
      __syncthreads();
    }
  }
  topkeys[(size_t)b * KPAD + tid] = sbuf[tid];
}

// ---------------------------------------------------------------------------
// Stage 3: greedy NMS + top-300 emission.  One WG (32 waves) per image.
// Phase A: parallel IoU>thr (j>i) bitmask via wave32 __ballot into 125 KB LDS.
// Phase B: single-wave sequential replay (1 mask word per lane, __shfl
//          broadcast of the alive bit) -- no block barriers in the loop.
// Then LDS scan ranks survivors; rank<300 rows written, rest zero-filled.
// ---------------------------------------------------------------------------
__global__ __launch_bounds__(1024) void nms_kernel(
    const float* __restrict__ x,
    const int* __restrict__ clsArr,
    const unsigned long long* __restrict__ topkeys,
    float* __restrict__ out) {
  __shared__ float    bo[KTOP * 4];     // class-offset boxes (for IoU)
  __shared__ float    area[KTOP];       // areas of offset boxes (matches ref)
  __shared__ float    scs[KTOP];
  __shared__ float    clf[KTOP];
  __shared__ unsigned sidx[KTOP];       // anchor index (for raw-box regather)
  __shared__ int      keep[KTOP];
  __shared__ int      scan[1024];
  extern __shared__ unsigned supMask[]; // KTOP*32 words = 128,000 B dynamic

  const int tid = threadIdx.x;
  const int b   = blockIdx.x;

  if (tid < KTOP) {
    unsigned long long key = topkeys[(size_t)b * KPAD + tid];
    unsigned hi  = (unsigned)(key >> 32);
    unsigned lo  = (unsigned)key;
    unsigned idx = 0xFFFFFFFFu - lo;
    unsigned fb  = (hi & 0x80000000u) ? (hi & 0x7FFFFFFFu) : ~hi;
    float score  = __uint_as_float(fb);  // exact round-trip (NEG if filtered)
    const float* r = x + ((size_t)b * N_ANC + idx) * ROW;
    float cx = r[0], cy = r[1], w = r[2], h = r[3];
    float x1 = cx - w * 0.5f, y1 = cy - h * 0.5f;
    float x2 = cx + w * 0.5f, y2 = cy + h * 0.5f;
    float c   = (float)clsArr[(size_t)b * N_ANC + idx];
    float off = c * MAXWH;
    float ox1 = x1 + off, oy1 = y1 + off, ox2 = x2 + off, oy2 = y2 + off;
    bo[tid * 4 + 0] = ox1; bo[tid * 4 + 1] = oy1;
    bo[tid * 4 + 2] = ox2; bo[tid * 4 + 3] = oy2;
    area[tid] = (ox2 - ox1) * (oy2 - oy1);   // same floats as reference
    scs[tid]  = score;
    clf[tid]  = c;
    sidx[tid] = idx;
    keep[tid] = (score > CONF) ? 1 : 0;      // keep0
  }
  __syncthreads();

  // -------- Phase A: suppression bitmask  (iou(i,j)>thr && j>i) -----------
  const int wid = tid >> 5, lane = tid & 31;
  for (int i = wid; i < KTOP; i += 32) {
    float ix1 = bo[i * 4 + 0], iy1 = bo[i * 4 + 1];
    float ix2 = bo[i * 4 + 2], iy2 = bo[i * 4 + 3];
    float ia  = area[i];
    for (int wq = 0; wq < 32; ++wq) {
      int j = wq * 32 + lane;
      bool sup = false;
      if (j < KTOP && j > i) {
        float xx1 = fmaxf(ix1, bo[j * 4 + 0]);
        float yy1 = fmaxf(iy1, bo[j * 4 + 1]);
        float xx2 = fminf(ix2, bo[j * 4 + 2]);
        float yy2 = fminf(iy2, bo[j * 4 + 3]);
        float inter = fmaxf(xx2 - xx1, 0.0f) * fmaxf(yy2 - yy1, 0.0f);
        float uni   = ia + area[j] - inter;
        sup = (inter / (uni + 1e-7f)) > IOU_T;
      }
      unsigned m = (unsigned)__ballot(sup);   // wave32 -> low 32 bits
      if (lane == 0) supMask[i * 32 + wq] = m;
    }
  }
  __syncthreads();

  // -------- Phase B: sequential greedy pass on wave 0 ---------------------
  if (wid == 0) {
    unsigned rem = 0;  // lane L owns removal bits for j in [L*32, L*32+32)
    for (int t = 0; t < 32; ++t) {
      int j = lane * 32 + t;
      if (j >= KTOP || !keep[j]) rem |= (1u << t);  // keep0==0 => never kept
    }
    for (int i = 0; i < KTOP; ++i) {
      unsigned ow = (unsigned)__shfl((int)rem, i >> 5);  // owner word of i
      if (((ow >> (i & 31)) & 1u) == 0u)                 // i still alive
        rem |= supMask[i * 32 + lane];                   // suppress its j>i
    }
    for (int t = 0; t < 32; ++t) {
      int j = lane * 32 + t;
      if (j < KTOP) keep[j] = ((rem >> t) & 1u) ? 0 : 1;
    }
  }
  __syncthreads();

  // -------- rank survivors (Hillis-Steele inclusive scan) -----------------
  int flag = (tid < KTOP) ? keep[tid] : 0;
  scan[tid] = flag;
  __syncthreads();
  for (int off2 = 1; off2 < 1024; off2 <<= 1) {
    int v   = scan[tid];
    int add = (tid >= off2) ? scan[tid - off2] : 0;
    __syncthreads();
    scan[tid] = v + add;
    __syncthreads();
  }
  int total  = scan[1023];
  int nvalid = total < MAXDET ? total : MAXDET;

  if (flag) {
    int rank = scan[tid] - 1;
    if (rank < MAXDET) {
      unsigned idx = sidx[tid];
      const float* r = x + ((size_t)b * N_ANC + idx) * ROW;
      float cx = r[0], cy = r[1], w = r[2], h = r[3];
      float* o = out + ((size_t)b * MAXDET + rank) * 6;
      o[0] = cx - w * 0.5f;
      o[1] = cy - h * 0.5f;
      o[2] = cx + w * 0.5f;
      o[3] = cy + h * 0.5f;
      o[4] = scs[tid];
      o[5] = clf[tid];
    }
  }
  // zero-fill invalid rows (reference multiplies by valid mask)
  for (int r2 = nvalid + tid; r2 < MAXDET; r2 += 1024) {
    float* o = out + ((size_t)b * MAXDET + r2) * 6;
    o[0] = o[1] = o[2] = o[3] = o[4] = o[5] = 0.0f;
  }
}

// ---------------------------------------------------------------------------
extern "C" void kernel_launch(void* const* d_in, const int* in_sizes, int n_in,
                              void* d_out, int out_size, void* d_ws,
                              size_t ws_size, hipStream_t stream) {
  (void)in_sizes; (void)n_in; (void)out_size; (void)ws_size;
  const float* x = (const float*)d_in[0];
  float* out = (float*)d_out;

  char* ws = (char*)d_ws;  // layout: keys (3.2MB) | cls (1.6MB) | topkeys (128KB)
  unsigned long long* keys    = (unsigned long long*)ws;
  int*                clsArr  = (int*)(ws + (size_t)B_IMG * N_ANC * 8);
  unsigned long long* topkeys =
      (unsigned long long*)(ws + (size_t)B_IMG * N_ANC * 12);

  const int nblk = (N_ANC + APB - 1) / APB;  // 197 blocks per image

  score_kernel<<<dim3(B_IMG * nblk), dim3(APB), 0, stream>>>(x, keys, clsArr);
  topk_kernel<<<dim3(B_IMG), dim3(1024), 0, stream>>>(keys, topkeys);
  nms_kernel<<<dim3(B_IMG), dim3(1024),
               (size_t)(KTOP * 32 * sizeof(unsigned)), stream>>>(
      x, clsArr, topkeys, out);
}